// RegionLoss_44787918963472
// MI455X (gfx1250) — compile-verified
//
#include <hip/hip_runtime.h>
#include <math.h>

// Problem constants (from reference)
#define NB 16
#define NT 50
#define NA 5
#define NC 8
#define NH 96
#define NW 96
#define CH (7 + NC)          // 15 channels per anchor
#define HW (NH * NW)         // 9216
#define NPB (NA * HW)        // 46080 predictions per batch
#define NTOT (NB * NPB)      // 737280 total predictions (== 2880 * 256 exactly)
#define RECSZ 12             // floats per target record
#define THRESH_F 0.6f
#define OBJ_W 100.0f         // OBJ^2 = 10^2 applied inside mse

typedef __attribute__((ext_vector_type(2))) float v2f;
typedef __attribute__((ext_vector_type(8))) float v8f;

__device__ __forceinline__ float sigmoidf_(float v) { return 1.0f / (1.0f + expf(-v)); }

__device__ __forceinline__ float iou_xywh(float px, float py, float pw, float pl,
                                          float gx, float gy, float gw, float gl) {
  float mx = fminf(px - pw * 0.5f, gx - gw * 0.5f);
  float Mx = fmaxf(px + pw * 0.5f, gx + gw * 0.5f);
  float my = fminf(py - pl * 0.5f, gy - gl * 0.5f);
  float My = fmaxf(py + pl * 0.5f, gy + gl * 0.5f);
  float cw = pw + gw - (Mx - mx);
  float chh = pl + gl - (My - my);
  float inter = (cw > 0.f && chh > 0.f) ? cw * chh : 0.f;
  return inter / (pw * pl + gw * gl - inter);
}

// Kernel 1: reset assignment map and the 3 output scalars every call.
__global__ void rl_init(int* __restrict__ assign, float* __restrict__ res) {
  int i = blockIdx.x * blockDim.x + threadIdx.x;
  if (i < NTOT) assign[i] = -1;
  if (i < 3) res[i] = 0.f;
}

// Kernel 2: per-batch sequential target processing (deterministic last-write-wins
// scatter in increasing-t order), target records, nGT / nCorrect.
__global__ void rl_targets(const float* __restrict__ out, const float* __restrict__ tgt,
                           const float* __restrict__ anc, float* __restrict__ rec,
                           int* __restrict__ assign, float* __restrict__ res) {
  int b = threadIdx.x;
  if (b >= NB) return;
  float validrun = 1.f;
  int nGT = 0, nCor = 0;
  for (int t = 0; t < NT; ++t) {
    const float* tp = tgt + ((size_t)b * NT + t) * 7;
    float t1 = tp[1];
    if (t1 == 0.f) validrun = 0.f;            // cumprod validity
    float* r = rec + ((size_t)b * NT + t) * RECSZ;
    if (validrun == 0.f) {
#pragma unroll
      for (int k = 0; k < RECSZ; ++k) r[k] = 0.f;
      continue;
    }
    float gx = t1 * (float)NW, gy = tp[2] * (float)NH;
    float gw = tp[3] * (float)NW, gl = tp[4] * (float)NH;
    // best anchor (first-wins ties, matches jnp.argmax)
    int bn = 0; float best = -1.f;
#pragma unroll
    for (int a2 = 0; a2 < NA; ++a2) {
      float aw2 = anc[2 * a2], ah2 = anc[2 * a2 + 1];
      float inter = fminf(gw, aw2) * fminf(gl, ah2);
      float u = inter / (gw * gl + aw2 * ah2 - inter);
      if (u > best) { best = u; bn = a2; }
    }
    int gi = (int)gx; gi = min(max(gi, 0), NW - 1);
    int gj = (int)gy; gj = min(max(gj, 0), NH - 1);
    float aw = anc[2 * bn], ah = anc[2 * bn + 1];
    r[0] = 1.f; r[1] = gx; r[2] = gy; r[3] = gw; r[4] = gl;
    r[5] = gx - (float)gi;  r[6] = gy - (float)gj;
    r[7] = logf(gw / aw);   r[8] = logf(gl / ah);
    r[9] = tp[5]; r[10] = tp[6]; r[11] = tp[0];
    int idx = ((b * NA + bn) * NH + gj) * NW + gi;
    assign[idx] = t;                          // later t overwrites earlier
    nGT++;
    // nCorrect: decode prediction at assigned cell, IoU vs target box
    const float* base = out + ((size_t)(b * NA + bn) * CH) * HW + (size_t)(gj * NW + gi);
    float px = sigmoidf_(base[0]) + (float)gi;
    float py = sigmoidf_(base[(size_t)1 * HW]) + (float)gj;
    float pw = expf(base[(size_t)2 * HW]) * aw;
    float pl = expf(base[(size_t)3 * HW]) * ah;
    if (iou_xywh(px, py, pw, pl, gx, gy, gw, gl) > 0.5f) nCor++;
  }
  atomicAdd(res + 1, (float)nCor);
  atomicAdd(res + 2, (float)nGT);
}

// Kernel 3: per-prediction loss, wave reduction via V_WMMA_F32_16X16X4_F32.
__global__ void __launch_bounds__(256) rl_main(const float* __restrict__ out,
                                               const float* __restrict__ anc,
                                               const float* __restrict__ rec,
                                               const int* __restrict__ assign,
                                               float* __restrict__ res) {
  __shared__ float sr[NT * RECSZ];
  __shared__ float wsum[8];
  const int tid = threadIdx.x;
  const int i = blockIdx.x * 256 + tid;
  const int b = i / NPB;                      // uniform within a block (NPB % 256 == 0)
  const float* rb = rec + (size_t)b * NT * RECSZ;
  for (int k = tid; k < NT * RECSZ; k += 256) sr[k] = rb[k];
  __syncthreads();

  const int rem  = i - b * NPB;
  const int a    = rem / HW;
  const int cell = rem - a * HW;
  const int jj   = cell / NW;
  const int ii   = cell - jj * NW;
  const float* base = out + ((size_t)(b * NA + a) * CH) * HW + (size_t)cell;

  // CDNA5 prefetch path: warm the class-logit cachelines (global_prefetch_b8)
#pragma unroll
  for (int c = 7; c < CH; ++c) __builtin_prefetch((const void*)(base + (size_t)c * HW), 0, 1);

  float ox  = base[0];
  float oy  = base[(size_t)1 * HW];
  float ow  = base[(size_t)2 * HW];
  float ol  = base[(size_t)3 * HW];
  float oim = base[(size_t)4 * HW];
  float ore = base[(size_t)5 * HW];
  float oc  = base[(size_t)6 * HW];

  float x = sigmoidf_(ox), y = sigmoidf_(oy), conf = sigmoidf_(oc);
  float aw = anc[2 * a], ah = anc[2 * a + 1];
  float px = x + (float)ii, py = y + (float)jj;
  float pw = expf(ow) * aw, pl = expf(ol) * ah;

  int asg = assign[i];
  float acc;
  if (asg >= 0) {
    const float* r = &sr[asg * RECSZ];
    float iou = iou_xywh(px, py, pw, pl, r[1], r[2], r[3], r[4]);  // tconf
    float dx = x - r[5], dy = y - r[6], dw = ow - r[7], dl = ol - r[8];
    float dim = oim - r[9], dre = ore - r[10], dc = conf - iou;
    // class cross-entropy on this cell's 8 logits
    float cl[NC];
#pragma unroll
    for (int c = 0; c < NC; ++c) cl[c] = base[(size_t)(7 + c) * HW];
    float m = cl[0];
#pragma unroll
    for (int c = 1; c < NC; ++c) m = fmaxf(m, cl[c]);
    float s = 0.f;
#pragma unroll
    for (int c = 0; c < NC; ++c) s += expf(cl[c] - m);
    float lse = m + logf(s);
    int ci = (int)r[11];
    float ce = lse - cl[ci];
    acc = dx * dx + dy * dy + dw * dw + dl * dl + dim * dim + dre * dre
        + OBJ_W * dc * dc + ce;
  } else {
    float maxiou = 0.f;
    for (int t = 0; t < NT; ++t) {
      const float* r = &sr[t * RECSZ];
      if (r[0] != 0.f)
        maxiou = fmaxf(maxiou, iou_xywh(px, py, pw, pl, r[1], r[2], r[3], r[4]));
    }
    acc = (maxiou > THRESH_F) ? 0.f : conf * conf;   // NOOBJ mask = 1
  }

  // --- wave32 reduction of acc via fp32 WMMA: D = A(16x4) * ones(4x16) + 0 ---
  // A layout: lanes 0-15 v[0] -> A[m][0], lanes 16-31 v[0] -> A[m][2]; v[1] = 0.
  // => D[m][*] = acc_m + acc_{m+16}. Lane n<16 holds D[0..7][n]; lane n>=16 holds
  // D[8..15][n-16]. Sum own 8, then shfl_xor(16) gives full wave sum.
  float tot;
#if __has_builtin(__builtin_amdgcn_wmma_f32_16x16x4_f32)
  {
    v2f av; av.x = acc;  av.y = 0.0f;
    v2f bv; bv.x = 1.0f; bv.y = 1.0f;
    v8f cv = {};
    cv = __builtin_amdgcn_wmma_f32_16x16x4_f32(false, av, false, bv, (short)0, cv,
                                               false, false);
    float s8 = cv[0] + cv[1] + cv[2] + cv[3] + cv[4] + cv[5] + cv[6] + cv[7];
    tot = s8 + __shfl_xor(s8, 16, 32);
  }
#else
  {
    tot = acc;
    for (int off = 16; off > 0; off >>= 1) tot += __shfl_xor(tot, off, 32);
  }
#endif

  int wid = tid >> 5;
  if ((tid & 31) == 0) wsum[wid] = tot;
  __syncthreads();
  if (tid == 0) {
    float t2 = 0.f;
#pragma unroll
    for (int k2 = 0; k2 < 8; ++k2) t2 += wsum[k2];
    atomicAdd(res, t2);
  }
}

extern "C" void kernel_launch(void* const* d_in, const int* in_sizes, int n_in,
                              void* d_out, int out_size, void* d_ws, size_t ws_size,
                              hipStream_t stream) {
  (void)in_sizes; (void)n_in; (void)out_size; (void)ws_size;
  const float* output  = (const float*)d_in[0];
  const float* target  = (const float*)d_in[1];
  const float* anchors = (const float*)d_in[2];
  float* res = (float*)d_out;                 // [loss, nCorrect, nGT] as f32
  int*   assign = (int*)d_ws;                                   // N ints
  float* rec = (float*)((char*)d_ws + (size_t)NTOT * sizeof(int)); // NB*NT*12 f32

  rl_init<<<(NTOT + 255) / 256, 256, 0, stream>>>(assign, res);
  rl_targets<<<1, 32, 0, stream>>>(output, target, anchors, rec, assign, res);
  rl_main<<<NTOT / 256, 256, 0, stream>>>(output, anchors, rec, assign, res);
}